// CapsuleLayer_21277267984738
// MI455X (gfx1250) — compile-verified
//
#include <hip/hip_runtime.h>
#include <hip/hip_bf16.h>

typedef __attribute__((ext_vector_type(2))) float v2f;
typedef __attribute__((ext_vector_type(4))) float v4f;
typedef __attribute__((ext_vector_type(8))) float v8f;

#define B_ 32
#define J_ 32
#define I_ 2048
#define T_ 16
#define D_ 16

// -----------------------------------------------------------------------------
// Kernel 1: u_hat[j][i][b][t] = sum_d W[j][i][t][d] * x[b][i][d]
// Per (j,i): U(16x32) = Wtile(16x16) * X_i^T(16x32) via v_wmma_f32_16x16x4_f32.
// One block per i; 8 waves x 4 j-values each. x-tile loaded once per wave.
// -----------------------------------------------------------------------------
__global__ __launch_bounds__(256) void uhat_gemm(const float* __restrict__ x,
                                                 const float* __restrict__ W,
                                                 float* __restrict__ u) {
    const int i    = blockIdx.x;        // 0..I_-1
    const int tid  = threadIdx.x;
    const int wave = tid >> 5;          // 0..7
    const int lane = tid & 31;
    const int n    = lane & 15;         // column within 16-wide tile
    const int hi   = lane >> 4;         // half-wave selector
    const int k0   = hi * 2;            // K rows {0,1} vs {2,3} per ISA layout

    // B operand: Xi^T chunks, Bx[c][h] covers d = 4c+k0..4c+k0+1 for b = n+16h
    v2f Bx[4][2];
#pragma unroll
    for (int h = 0; h < 2; ++h) {
        const float* xp = x + ((size_t)(n + 16 * h) * I_ + i) * D_;
#pragma unroll
        for (int c = 0; c < 4; ++c)
            Bx[c][h] = *(const v2f*)(xp + 4 * c + k0);
    }

#pragma unroll
    for (int jj = 0; jj < 4; ++jj) {
        const int j = (wave << 2) + jj;
        // A operand: W[j][i][t=n][d], lanes 0-15 rows with K={0,1}, 16-31 K={2,3}
        const float* wp = W + (((size_t)j * I_ + i) * T_ + n) * D_;
        v2f A[4];
#pragma unroll
        for (int c = 0; c < 4; ++c)
            A[c] = *(const v2f*)(wp + 4 * c + k0);

        v8f acc0 = {0.f, 0.f, 0.f, 0.f, 0.f, 0.f, 0.f, 0.f};
        v8f acc1 = acc0;
#pragma unroll
        for (int c = 0; c < 4; ++c) {
            acc0 = __builtin_amdgcn_wmma_f32_16x16x4_f32(
                false, A[c], false, Bx[c][0], (short)0, acc0, false, false);
            acc1 = __builtin_amdgcn_wmma_f32_16x16x4_f32(
                false, A[c], false, Bx[c][1], (short)0, acc1, false, false);
        }

        // D layout: lane holds b = n+16h, t = 8*hi + r (r = vgpr index 0..7)
#pragma unroll
        for (int h = 0; h < 2; ++h) {
            const v8f acc = h ? acc1 : acc0;
            float* up = u + (((size_t)j * I_ + i) * B_ + (n + 16 * h)) * T_ + 8 * hi;
            v4f lo = __builtin_shufflevector(acc, acc, 0, 1, 2, 3);
            v4f hv = __builtin_shufflevector(acc, acc, 4, 5, 6, 7);
            *(v4f*)(up)     = lo;
            *(v4f*)(up + 4) = hv;
        }
    }
}

// -----------------------------------------------------------------------------
// Kernel 2: c[b][j][i] = softmax_j(b_ij[b][j][i]); one thread per (b,i)
// -----------------------------------------------------------------------------
__global__ __launch_bounds__(256) void softmax_j(const float* __restrict__ bij,
                                                 float* __restrict__ c) {
    const int g = blockIdx.x * 256 + threadIdx.x;   // 0..B_*I_-1
    const int b = g >> 11;                          // / I_
    const int i = g & (I_ - 1);
    const float* bp = bij + (size_t)b * J_ * I_ + i;
    float vals[J_];
    float m = -3.4e38f;
#pragma unroll
    for (int j = 0; j < J_; ++j) {
        vals[j] = bp[(size_t)j * I_];
        m = fmaxf(m, vals[j]);
    }
    float s = 0.f;
#pragma unroll
    for (int j = 0; j < J_; ++j) {
        vals[j] = __expf(vals[j] - m);
        s += vals[j];
    }
    const float inv = 1.0f / s;
    float* cp = c + (size_t)b * J_ * I_ + i;
#pragma unroll
    for (int j = 0; j < J_; ++j)
        cp[(size_t)j * I_] = vals[j] * inv;
}

// -----------------------------------------------------------------------------
// Kernel 3: s[b][j][t] = sum_i c[b][j][i] * u[j][i][b][t];  v = squash(s)
// One block per (b,j); thread (grp,t): grp strides i, t is element.
// uniform!=0 -> c == 1/32 everywhere (iteration 0).
// -----------------------------------------------------------------------------
__global__ __launch_bounds__(256) void sv_kernel(const float* __restrict__ u,
                                                 const float* __restrict__ c,
                                                 float* __restrict__ v,
                                                 int uniform) {
    const int bj  = blockIdx.x;         // 0..B_*J_-1
    const int b   = bj >> 5;
    const int j   = bj & 31;
    const int tid = threadIdx.x;
    const int t   = tid & 15;
    const int grp = tid >> 4;           // 16 i-groups

    const float* ubase = u + (size_t)j * I_ * (B_ * T_) + (size_t)b * T_ + t;
    const float* cbase = c + ((size_t)b * J_ + j) * I_;

    float acc = 0.f;
    for (int i = grp; i < I_; i += 16) {
        const float uv = ubase[(size_t)i * (B_ * T_)];
        acc += uniform ? uv : uv * cbase[i];
    }

    __shared__ float red[256];
    __shared__ float scale_s;
    red[tid] = acc;
    __syncthreads();
    for (int sft = 128; sft >= 16; sft >>= 1) {
        if (tid < sft) red[tid] += red[tid + sft];
        __syncthreads();
    }
    if (tid == 0) {
        float sq = 0.f;
#pragma unroll
        for (int tt = 0; tt < 16; ++tt) { const float sv = red[tt]; sq += sv * sv; }
        if (uniform) sq *= (1.0f / 32.f) * (1.0f / 32.f);
        scale_s = sq / (1.0f + sq) / sqrtf(sq + 1e-9f);
    }
    __syncthreads();
    if (tid < 16) {
        float sv = red[t];
        if (uniform) sv *= (1.0f / 32.f);
        v[((size_t)b * J_ + j) * T_ + t] = scale_s * sv;
    }
}

// -----------------------------------------------------------------------------
// Kernel 4: b_ij[b][j][i] (+)= sum_t u[j][i][b][t] * v[b][j][t]
// i innermost across threads -> coalesced b_ij stores; u reads hit L2.
// -----------------------------------------------------------------------------
__global__ __launch_bounds__(256) void bupd_kernel(const float* __restrict__ u,
                                                   const float* __restrict__ v,
                                                   float* __restrict__ bij,
                                                   int first) {
    const size_t g = (size_t)blockIdx.x * 256 + threadIdx.x;   // 0..B_*J_*I_-1
    const int i  = (int)(g & (I_ - 1));
    const int bj = (int)(g >> 11);
    const int j  = bj & 31;
    const int b  = bj >> 5;

    const float* up = u + (((size_t)j * I_ + i) * B_ + b) * T_;
    const float* vp = v + ((size_t)b * J_ + j) * T_;

    float dot = 0.f;
#pragma unroll
    for (int q = 0; q < 4; ++q) {
        const v4f uv = *(const v4f*)(up + 4 * q);
        const v4f vv = *(const v4f*)(vp + 4 * q);
        dot += uv.x * vv.x + uv.y * vv.y + uv.z * vv.z + uv.w * vv.w;
    }

    const size_t bidx = ((size_t)b * J_ + j) * I_ + i;
    bij[bidx] = first ? dot : (bij[bidx] + dot);
}

// -----------------------------------------------------------------------------
extern "C" void kernel_launch(void* const* d_in, const int* in_sizes, int n_in,
                              void* d_out, int out_size, void* d_ws, size_t ws_size,
                              hipStream_t stream) {
    const float* x = (const float*)d_in[0];   // [B, I, D]
    const float* W = (const float*)d_in[1];   // [1, J, I, T, D]
    float* out = (float*)d_out;               // [B, J, T] — doubles as v buffer

    float* u   = (float*)d_ws;                                 // [J][I][B][T]
    float* bij = u + (size_t)J_ * I_ * B_ * T_;                 // [B][J][I]
    float* c   = bij + (size_t)B_ * J_ * I_;                    // [B][J][I]

    uhat_gemm<<<I_, 256, 0, stream>>>(x, W, u);

    // r = 0: softmax(0) == 1/J (uniform), b_ij update overwrites (first=1)
    sv_kernel<<<B_ * J_, 256, 0, stream>>>(u, c, out, 1);
    bupd_kernel<<<(B_ * J_ * I_) / 256, 256, 0, stream>>>(u, out, bij, 1);

    // r = 1
    softmax_j<<<(B_ * I_) / 256, 256, 0, stream>>>(bij, c);
    sv_kernel<<<B_ * J_, 256, 0, stream>>>(u, c, out, 0);
    bupd_kernel<<<(B_ * J_ * I_) / 256, 256, 0, stream>>>(u, out, bij, 0);

    // r = 2 (final v -> d_out)
    softmax_j<<<(B_ * I_) / 256, 256, 0, stream>>>(bij, c);
    sv_kernel<<<B_ * J_, 256, 0, stream>>>(u, c, out, 0);
}